// CoxLoss_56642028700187
// MI455X (gfx1250) — compile-verified
//
#include <hip/hip_runtime.h>
#include <hip/hip_bf16.h>
#include <math.h>

// Cox partial likelihood (Breslow ties, mean reduction) for integer durations
// in [0, 365). Sort-free formulation:
//   S_d  = sum_{dur=d} exp(clip(pred,-20,20))        (365-bucket histogram)
//   R_d  = sum_{d'>=d} S_d'                          (suffix sum)
//   ll   = sum_i e_i*y_i - sum_d m_d*log(max(R_d,1e-12))
//   loss = -ll / max(sum_e, 1)
// Degenerate branch (sum_e == 0): e -> 1e-8 everywhere, needs sum(y) and
// per-duration total counts c_d; n_events = max(1e-8*n, 1).
//
// Memory-bound: 48 MB streamed -> ~2.1 us roofline at 23.3 TB/s. WMMA has no
// role (no matrix op); CDNA5 paths exercised: wave32, b128 global loads,
// ds_add_u64/u32 LDS atomics, global_prefetch_b8.
// All cross-thread reductions use fixed-point integer atomics -> bitwise
// deterministic across graph replays.

#define NDUR 365
#define NB   368   // padded bucket count (keeps 8-byte alignment of ws layout)

#define EY_SCALE   65536.0f        // 2^16 fixed point for exp sums
#define EY_INV     (1.0 / 65536.0)
#define Y_SCALE    16777216.0f     // 2^24 fixed point for y sums
#define Y_INV      (1.0 / 16777216.0)

__global__ __launch_bounds__(512) void cox_zero(unsigned long long* __restrict__ S64,
                                                int* __restrict__ M,
                                                int* __restrict__ C,
                                                unsigned long long* __restrict__ sc64,
                                                int* __restrict__ sE) {
  int tid = threadIdx.x;
  for (int i = tid; i < NB; i += blockDim.x) { S64[i] = 0ull; M[i] = 0; C[i] = 0; }
  if (tid == 0) { sc64[0] = 0ull; sc64[1] = 0ull; *sE = 0; }
}

__global__ __launch_bounds__(256) void cox_hist(const float* __restrict__ pred,
                                                const int* __restrict__ dur,
                                                const int* __restrict__ ev,
                                                int n,
                                                unsigned long long* __restrict__ S64,
                                                int* __restrict__ M,
                                                int* __restrict__ C,
                                                unsigned long long* __restrict__ sc64,
                                                int* __restrict__ sE) {
  __shared__ unsigned long long ls[NDUR];   // fixed-point exp sums
  __shared__ int lm[NDUR];                  // event counts
  __shared__ int lc[NDUR];                  // total counts
  __shared__ unsigned long long l_ey;       // fixed-point sum(e*y)
  __shared__ unsigned long long l_y;        // fixed-point sum(y)
  __shared__ int l_e;                       // event count

  const int tid = threadIdx.x;
  for (int i = tid; i < NDUR; i += blockDim.x) { ls[i] = 0ull; lm[i] = 0; lc[i] = 0; }
  if (tid == 0) { l_ey = 0ull; l_y = 0ull; l_e = 0; }
  __syncthreads();

  float my_ey = 0.0f;  // sum of y over events (this thread)
  float my_y  = 0.0f;  // sum of y (this thread)
  int   my_e  = 0;     // event count (this thread)

  const int nvec   = n >> 2;
  const int stride = gridDim.x * blockDim.x;
  const float4* __restrict__ p4 = (const float4*)pred;
  const int4*  __restrict__ d4 = (const int4*)dur;
  const int4*  __restrict__ e4 = (const int4*)ev;

  for (int v = blockIdx.x * blockDim.x + tid; v < nvec; v += stride) {
    // prefetch the next grid-stride tile (speculative; OOB drops silently)
    __builtin_prefetch(p4 + v + stride, 0, 0);
    __builtin_prefetch(d4 + v + stride, 0, 0);
    __builtin_prefetch(e4 + v + stride, 0, 0);

    float4 p = p4[v];
    int4   d = d4[v];
    int4   e = e4[v];
    float ys[4] = {p.x, p.y, p.z, p.w};
    int   dd[4] = {d.x, d.y, d.z, d.w};
    int   ee[4] = {e.x, e.y, e.z, e.w};
#pragma unroll
    for (int k = 0; k < 4; ++k) {
      float y  = ys[k];
      float yc = fminf(fmaxf(y, -20.0f), 20.0f);
      float eyv = expf(yc);
      unsigned long long fx = (unsigned long long)(eyv * EY_SCALE + 0.5f);
      int b = dd[k];
      if (b < 0) b = 0;
      if (b >= NDUR) b = NDUR - 1;            // memory safety; inputs are in-range
      atomicAdd(&ls[b], fx);                  // ds_add_u64
      atomicAdd(&lc[b], 1);                   // ds_add_u32
      if (ee[k]) { my_e++; my_ey += y; atomicAdd(&lm[b], 1); }
      my_y += y;
    }
  }

  // scalar tail (n % 4 != 0); no-op for N = 4,194,304
  for (int i = (nvec << 2) + blockIdx.x * blockDim.x + tid; i < n; i += stride) {
    float y  = pred[i];
    float yc = fminf(fmaxf(y, -20.0f), 20.0f);
    unsigned long long fx = (unsigned long long)(expf(yc) * EY_SCALE + 0.5f);
    int b = dur[i];
    if (b < 0) b = 0;
    if (b >= NDUR) b = NDUR - 1;
    atomicAdd(&ls[b], fx);
    atomicAdd(&lc[b], 1);
    if (ev[i]) { my_e++; my_ey += y; atomicAdd(&lm[b], 1); }
    my_y += y;
  }

  // fold per-thread scalars into block scalars (integer -> order-independent)
  long long fx_ey = llrintf(my_ey * Y_SCALE);
  long long fx_y  = llrintf(my_y  * Y_SCALE);
  atomicAdd(&l_ey, (unsigned long long)fx_ey);
  atomicAdd(&l_y,  (unsigned long long)fx_y);
  atomicAdd(&l_e,  my_e);
  __syncthreads();

  // flush block-local histogram to global
  for (int i = tid; i < NDUR; i += blockDim.x) {
    unsigned long long s = ls[i];
    int m = lm[i], c = lc[i];
    if (s) atomicAdd(&S64[i], s);
    if (m) atomicAdd(&M[i], m);
    if (c) atomicAdd(&C[i], c);
  }
  if (tid == 0) {
    atomicAdd(&sc64[0], l_ey);
    atomicAdd(&sc64[1], l_y);
    atomicAdd(sE, l_e);
  }
}

__global__ __launch_bounds__(512) void cox_final(const unsigned long long* __restrict__ S64,
                                                 const int* __restrict__ M,
                                                 const int* __restrict__ C,
                                                 const unsigned long long* __restrict__ sc64,
                                                 const int* __restrict__ sE,
                                                 float* __restrict__ out,
                                                 int n) {
  __shared__ double dR[NDUR];
  __shared__ double redM[512];
  __shared__ double redC[512];
  const int tid = threadIdx.x;

  if (tid < NDUR) dR[tid] = (double)S64[tid] * EY_INV;
  __syncthreads();

  // inclusive suffix sum over 365 buckets (serial in double; trivial cost)
  if (tid == 0) {
    for (int d = NDUR - 2; d >= 0; --d) dR[d] += dR[d + 1];
  }
  __syncthreads();

  double pM = 0.0, pC = 0.0;
  if (tid < NDUR) {
    double r = dR[tid];
    if (r < 1e-12) r = 1e-12;
    double lg = log(r);
    pM = (double)M[tid] * lg;
    pC = (double)C[tid] * lg;
  }
  redM[tid] = pM;
  redC[tid] = pC;
  __syncthreads();
  for (int s = 256; s > 0; s >>= 1) {
    if (tid < s) { redM[tid] += redM[tid + s]; redC[tid] += redC[tid + s]; }
    __syncthreads();
  }

  if (tid == 0) {
    double sum_ey = (double)(long long)sc64[0] * Y_INV;
    double sumY   = (double)(long long)sc64[1] * Y_INV;
    int sumE = *sE;
    double loss;
    if (sumE > 0) {
      double total = sum_ey - redM[0];
      double nev = (double)sumE;
      if (nev < 1.0) nev = 1.0;
      loss = -total / nev;
    } else {
      // reference: e -> e + 1e-8 everywhere when no events
      double total = 1e-8 * (sumY - redC[0]);
      double nev = 1e-8 * (double)n;
      if (nev < 1.0) nev = 1.0;
      loss = -total / nev;
    }
    out[0] = (float)loss;
  }
}

extern "C" void kernel_launch(void* const* d_in, const int* in_sizes, int n_in,
                              void* d_out, int out_size, void* d_ws, size_t ws_size,
                              hipStream_t stream) {
  const float* pred = (const float*)d_in[0];
  const int*   dur  = (const int*)d_in[1];
  const int*   ev   = (const int*)d_in[2];
  float*       out  = (float*)d_out;
  const int n = in_sizes[0];

  // workspace layout (all 8-byte aligned; total ~5.9 KB)
  unsigned long long* S64  = (unsigned long long*)d_ws;       // NB u64
  int*                M    = (int*)(S64 + NB);                // NB i32
  int*                C    = M + NB;                          // NB i32
  unsigned long long* sc64 = (unsigned long long*)(C + NB);   // 2 u64 scalars
  int*                sE   = (int*)(sc64 + 2);                // 1 i32

  cox_zero<<<1, 512, 0, stream>>>(S64, M, C, sc64, sE);

  int blocks = 1024;                       // 8 waves/block; limits same-address
  int maxb = (n / 4 + 255) / 256;          // global-atomic chains to ~1024/bucket
  if (maxb < 1) maxb = 1;
  if (blocks > maxb) blocks = maxb;
  cox_hist<<<blocks, 256, 0, stream>>>(pred, dur, ev, n, S64, M, C, sc64, sE);

  cox_final<<<1, 512, 0, stream>>>(S64, M, C, sc64, sE, out, n);
}